// CombineGraph_57878979281132
// MI455X (gfx1250) — compile-verified
//
#include <hip/hip_runtime.h>
#include <math.h>

typedef __attribute__((ext_vector_type(2))) float v2f;
typedef __attribute__((ext_vector_type(8))) float v8f;

#define D     100
#define B     512
#define S     50
#define NS    12
#define LASTL 3
#define NADJ  12
#define BS    (B*S)
#define NPAD  112      /* 7 tiles of 16 covering N=100 */
#define SCALE 0.1f     /* 1/sqrt(100) */

__device__ inline float wave_sum(float v) {
#pragma unroll
  for (int off = 16; off > 0; off >>= 1) v += __shfl_xor(v, off, 32);
  return v;
}

// ---------------- gather rows of embedding ----------------
__global__ void gather_rows(const float* __restrict__ emb, const int* __restrict__ idx,
                            float* __restrict__ out, int n) {
  int i = blockIdx.x * blockDim.x + threadIdx.x;
  if (i >= n) return;
  int r = i / D, c = i - r * D;
  out[i] = emb[(long)idx[r] * D + c];
}

// ---------------- star = masked mean over S ----------------
__global__ void star_kernel(const float* __restrict__ h, const int* __restrict__ inputs,
                            float* __restrict__ star) {
  int b = blockIdx.x;
  int t = threadIdx.x;
  __shared__ float s_den;
  if (t == 0) {
    int c = 0;
    for (int s = 0; s < S; ++s) c += (inputs[b*S + s] != 0) ? 1 : 0;
    s_den = (float)(c > 0 ? c : 1);
  }
  __syncthreads();
  float denom = s_den;
  for (int d = t; d < D; d += blockDim.x) {
    float acc = 0.f;
    for (int s = 0; s < S; ++s)
      if (inputs[b*S + s] != 0) acc += h[(b*S + s)*D + d];
    star[b*D + d] = acc / denom;
  }
}

// ---------------- pad+transpose weight: WT[c][k] = (c<100) ? W[k][c] : 0 ----------------
__global__ void pad_weightT(const float* __restrict__ W, float* __restrict__ WT, int K) {
  int i = blockIdx.x * blockDim.x + threadIdx.x;
  if (i >= NPAD * K) return;
  int c = i / K, k = i - c * K;
  WT[i] = (c < D) ? W[k * D + c] : 0.f;
}

// ---------------- zero-pad adj: adjP[b][r<64][k<52] ----------------
__global__ void pad_adj_kernel(const float* __restrict__ adj, float* __restrict__ adjP) {
  int i = blockIdx.x * blockDim.x + threadIdx.x;
  if (i >= B * 64 * 52) return;
  int b = i / (64 * 52), rem = i - b * 64 * 52;
  int r = rem / 52, k = rem - r * 52;
  adjP[i] = (r < S && k < S) ? adj[(long)b*S*S + r*S + k] : 0.f;
}

// ---------------- pad+transpose per-batch X: XT[b][c<112][k<52] = X[b][k][c] ----------------
__global__ void padT_kernel(const float* __restrict__ X, float* __restrict__ XT) {
  int i = blockIdx.x * blockDim.x + threadIdx.x;
  if (i >= B * NPAD * 52) return;
  int b = i / (NPAD * 52), rem = i - b * NPAD * 52;
  int c = rem / 52, k = rem - c * 52;
  XT[i] = (c < D && k < S) ? X[(long)b*S*D + k*D + c] : 0.f;
}

// ---------------- f32 WMMA GEMM: C[M,100] = act(A[M,K] @ W)  (WT pre-padded/transposed)
// M % 16 == 0, K % 4 == 0.  Inner loop: 2x global_load_b64 + 1x v_wmma, no branches.
template<int K>
__global__ void wmma_gemm(const float* __restrict__ A, const float* __restrict__ WT,
                          float* __restrict__ C, int M, int relu) {
  constexpr int TN = NPAD / 16;                 // 7
  int wid  = (blockIdx.x * blockDim.x + threadIdx.x) >> 5;
  int lane = threadIdx.x & 31;
  int tilesM = M >> 4;
  if (wid >= tilesM * TN) return;               // wave-uniform: EXEC stays full
  int tm = wid / TN, tn = wid - tm * TN;
  int half = lane >> 4, lid = lane & 15;
  int row = tm * 16 + lid;
  int col = tn * 16 + lid;
  const float* Ar = A  + (long)row * K + half * 2;
  const float* Br = WT + (long)col * K + half * 2;
  v8f acc = {0.f,0.f,0.f,0.f,0.f,0.f,0.f,0.f};
#pragma unroll
  for (int kb = 0; kb < K; kb += 4) {
    v2f af = *(const v2f*)(Ar + kb);
    v2f bf = *(const v2f*)(Br + kb);
    acc = __builtin_amdgcn_wmma_f32_16x16x4_f32(false, af, false, bf,
                                                (short)0, acc, false, false);
  }
  if (col < D) {
#pragma unroll
    for (int r = 0; r < 8; ++r) {
      float v = acc[r];
      if (relu) v = fmaxf(v, 0.f);
      C[(long)(tm*16 + half*8 + r) * D + col] = v;
    }
  }
}

// ---------------- dual GEMM: C1 = A@W1, C2 = A@W2 (shared A fragments) ----------------
template<int K>
__global__ void wmma_gemm_dual(const float* __restrict__ A,
                               const float* __restrict__ W1T, const float* __restrict__ W2T,
                               float* __restrict__ C1, float* __restrict__ C2, int M) {
  constexpr int TN = NPAD / 16;
  int wid  = (blockIdx.x * blockDim.x + threadIdx.x) >> 5;
  int lane = threadIdx.x & 31;
  int tilesM = M >> 4;
  if (wid >= tilesM * TN) return;
  int tm = wid / TN, tn = wid - tm * TN;
  int half = lane >> 4, lid = lane & 15;
  int row = tm * 16 + lid;
  int col = tn * 16 + lid;
  const float* Ar  = A   + (long)row * K + half * 2;
  const float* B1r = W1T + (long)col * K + half * 2;
  const float* B2r = W2T + (long)col * K + half * 2;
  v8f acc1 = {0.f,0.f,0.f,0.f,0.f,0.f,0.f,0.f};
  v8f acc2 = {0.f,0.f,0.f,0.f,0.f,0.f,0.f,0.f};
#pragma unroll
  for (int kb = 0; kb < K; kb += 4) {
    v2f af  = *(const v2f*)(Ar  + kb);
    v2f bf1 = *(const v2f*)(B1r + kb);
    v2f bf2 = *(const v2f*)(B2r + kb);
    acc1 = __builtin_amdgcn_wmma_f32_16x16x4_f32(false, af, false, bf1,
                                                 (short)0, acc1, false, false);
    acc2 = __builtin_amdgcn_wmma_f32_16x16x4_f32(false, af, false, bf2,
                                                 (short)0, acc2, false, false);
  }
  if (col < D) {
#pragma unroll
    for (int r = 0; r < 8; ++r) {
      long o = (long)(tm*16 + half*8 + r) * D + col;
      C1[o] = acc1[r];
      C2[o] = acc2[r];
    }
  }
}

// ---------------- concat GEMM: C = relu([A0 | A1] @ Wg), K = 200 ----------------
__global__ void wmma_gemm_cat(const float* __restrict__ A0, const float* __restrict__ A1,
                              const float* __restrict__ WT,   // [112][200]
                              float* __restrict__ C, int M) {
  constexpr int TN = NPAD / 16, Kh = D, K = 2 * D;
  int wid  = (blockIdx.x * blockDim.x + threadIdx.x) >> 5;
  int lane = threadIdx.x & 31;
  int tilesM = M >> 4;
  if (wid >= tilesM * TN) return;
  int tm = wid / TN, tn = wid - tm * TN;
  int half = lane >> 4, lid = lane & 15;
  int row = tm * 16 + lid;
  int col = tn * 16 + lid;
  const float* A0r = A0 + (long)row * Kh + half * 2;
  const float* A1r = A1 + (long)row * Kh + half * 2;
  const float* Br  = WT + (long)col * K  + half * 2;
  v8f acc = {0.f,0.f,0.f,0.f,0.f,0.f,0.f,0.f};
#pragma unroll
  for (int kb = 0; kb < Kh; kb += 4) {
    v2f af = *(const v2f*)(A0r + kb);
    v2f bf = *(const v2f*)(Br + kb);
    acc = __builtin_amdgcn_wmma_f32_16x16x4_f32(false, af, false, bf,
                                                (short)0, acc, false, false);
  }
#pragma unroll
  for (int kb = 0; kb < Kh; kb += 4) {
    v2f af = *(const v2f*)(A1r + kb);
    v2f bf = *(const v2f*)(Br + Kh + kb);
    acc = __builtin_amdgcn_wmma_f32_16x16x4_f32(false, af, false, bf,
                                                (short)0, acc, false, false);
  }
  if (col < D) {
#pragma unroll
    for (int r = 0; r < 8; ++r)
      C[(long)(tm*16 + half*8 + r) * D + col] = fmaxf(acc[r], 0.f);
  }
}

// ---------------- batched h_n[b] = adj[b] @ X[b] via padded buffers ----------------
__global__ void adj_gemm(const float* __restrict__ adjP,   // [B][64][52]
                         const float* __restrict__ XT,     // [B][112][52]
                         float* __restrict__ Y) {           // [B][50][100]
  constexpr int TM = 4, TN = NPAD / 16, PB = TM * TN, KP = 52;
  int wid  = (blockIdx.x * blockDim.x + threadIdx.x) >> 5;
  int lane = threadIdx.x & 31;
  if (wid >= B * PB) return;
  int b = wid / PB, rem = wid - b * PB;
  int tm = rem / TN, tn = rem - tm * TN;
  int half = lane >> 4, lid = lane & 15;
  int row = tm * 16 + lid;
  int col = tn * 16 + lid;
  const float* Ar = adjP + (long)b*64*KP   + (long)row * KP + half * 2;
  const float* Br = XT   + (long)b*NPAD*KP + (long)col * KP + half * 2;
  v8f acc = {0.f,0.f,0.f,0.f,0.f,0.f,0.f,0.f};
#pragma unroll
  for (int kb = 0; kb < KP; kb += 4) {
    v2f af = *(const v2f*)(Ar + kb);
    v2f bf = *(const v2f*)(Br + kb);
    acc = __builtin_amdgcn_wmma_f32_16x16x4_f32(false, af, false, bf,
                                                (short)0, acc, false, false);
  }
  if (col < D) {
#pragma unroll
    for (int r = 0; r < 8; ++r) {
      int orow = tm*16 + half*8 + r;
      if (orow < S) Y[(long)b*S*D + (long)orow*D + col] = acc[r];
    }
  }
}

// ---------------- gate = sigmoid((hq . sk)*SCALE);  hl = (1-g)*h_n + g*star ----------------
__global__ void gate_hl_kernel(const float* __restrict__ hq, const float* __restrict__ sk,
                               const float* hn, const float* __restrict__ star,
                               float* out) {                 // out may alias hn (in-place)
  int wid  = (blockIdx.x * blockDim.x + threadIdx.x) >> 5;   // (b,s)
  int lane = threadIdx.x & 31;
  if (wid >= BS) return;
  int b = wid / S;
  const float* hqr = hq + (long)wid * D;
  const float* skr = sk + b * D;
  float part = 0.f;
  for (int d = lane; d < D; d += 32) part += hqr[d] * skr[d];
  float dot  = wave_sum(part);
  float gate = 1.f / (1.f + expf(-dot * SCALE));
  const float* sr = star + b * D;
  for (int d = lane; d < D; d += 32) {
    float v = hn[(long)wid * D + d];
    out[(long)wid * D + d] = (1.f - gate) * v + gate * sr[d];
  }
}

// ---------------- global branch: agg = softmax(num_w) . embedding[adj_all] ----------------
__global__ void agg_kernel(const float* __restrict__ emb, const int* __restrict__ inputs,
                           const int* __restrict__ adj_all, const float* __restrict__ num_w,
                           float* __restrict__ agg) {
  int wid  = (blockIdx.x * blockDim.x + threadIdx.x) >> 5;   // (b,s)
  int lane = threadIdx.x & 31;
  if (wid >= BS) return;
  long id = inputs[wid];
  float w[NS]; int nid[NS];
  float m = -1e30f;
#pragma unroll
  for (int n = 0; n < NS; ++n) {
    nid[n] = adj_all[id * NS + n];
    w[n]   = num_w [id * NS + n];
    m = fmaxf(m, w[n]);
  }
  float ssum = 0.f;
#pragma unroll
  for (int n = 0; n < NS; ++n) { w[n] = expf(w[n] - m); ssum += w[n]; }
  float inv = 1.f / ssum;
  for (int d = lane; d < D; d += 32) {
    float acc = 0.f;
#pragma unroll
    for (int n = 0; n < NS; ++n) acc += w[n] * emb[(long)nid[n] * D + d];
    agg[(long)wid * D + d] = acc * inv;
  }
}

// ---------------- output = s0*l2norm(hl) + s1*l2norm(hg) ----------------
__global__ void combine_kernel(const float* __restrict__ hl, const float* __restrict__ hg,
                               const float* __restrict__ svec, float* __restrict__ out) {
  int wid  = (blockIdx.x * blockDim.x + threadIdx.x) >> 5;   // (b,s)
  int lane = threadIdx.x & 31;
  if (wid >= BS) return;
  const float* a = hl + (long)wid * D;
  const float* g = hg + (long)wid * D;
  float pa = 0.f, pg = 0.f;
  for (int d = lane; d < D; d += 32) { pa += a[d]*a[d]; pg += g[d]*g[d]; }
  float na = fmaxf(sqrtf(wave_sum(pa)), 1e-12f);
  float ng = fmaxf(sqrtf(wave_sum(pg)), 1e-12f);
  float s0 = svec[0], s1 = svec[1];
  for (int d = lane; d < D; d += 32)
    out[(long)wid * D + d] = s0 * a[d] / na + s1 * g[d] / ng;
}

// ---------------- LastAggregator: attention over NADJ, mean over LASTL ----------------
__global__ void last_kernel(const float* __restrict__ emb, const float* __restrict__ item_h,
                            const float* __restrict__ q, const int* __restrict__ adj_items,
                            float* __restrict__ out) {
  int wid  = (blockIdx.x * blockDim.x + threadIdx.x) >> 5;   // b
  int lane = threadIdx.x & 31;
  if (wid >= B) return;
  int b = wid;
  float lh[4] = {0.f,0.f,0.f,0.f};
  for (int l = 0; l < LASTL; ++l) {
    int aid[NADJ];
#pragma unroll
    for (int n = 0; n < NADJ; ++n) aid[n] = adj_items[b*(LASTL*NADJ) + l*NADJ + n];
    const float* qr = q + (long)(b*LASTL + l) * D;
    float att[NADJ]; float m = -1e30f;
#pragma unroll
    for (int n = 0; n < NADJ; ++n) {
      float part = 0.f;
      for (int d = lane; d < D; d += 32) part += qr[d] * emb[(long)aid[n]*D + d];
      att[n] = wave_sum(part) * SCALE;
      m = fmaxf(m, att[n]);
    }
    float ssum = 0.f;
#pragma unroll
    for (int n = 0; n < NADJ; ++n) { att[n] = expf(att[n] - m); ssum += att[n]; }
    float inv = 1.f / ssum;
    const float* ih = item_h + (long)(b*LASTL + l) * D;
#pragma unroll
    for (int c = 0; c < 4; ++c) {
      int d = lane + 32*c;
      if (d < D) {
        float v = ih[d];
#pragma unroll
        for (int n = 0; n < NADJ; ++n) v += att[n] * inv * emb[(long)aid[n]*D + d];
        lh[c] += v;
      }
    }
  }
#pragma unroll
  for (int c = 0; c < 4; ++c) {
    int d = lane + 32*c;
    if (d < D) out[b*D + d] = lh[c] * (1.f/3.f);
  }
}

extern "C" void kernel_launch(void* const* d_in, const int* in_sizes, int n_in,
                              void* d_out, int out_size, void* d_ws, size_t ws_size,
                              hipStream_t stream) {
  (void)in_sizes; (void)n_in; (void)out_size; (void)ws_size;
  const int*   inputs     = (const int*)  d_in[0];
  const float* adj        = (const float*)d_in[1];
  /* d_in[2] = item : unused by the reference */
  const int*   last_items = (const int*)  d_in[3];
  const int*   adj_items  = (const int*)  d_in[4];
  const float* embedding  = (const float*)d_in[5];
  const float* svec       = (const float*)d_in[6];
  const float* W_star_a   = (const float*)d_in[7];
  const float* W_star_q   = (const float*)d_in[8];
  const float* W_star_k   = (const float*)d_in[9];
  const float* W_g        = (const float*)d_in[10];
  const float* W_l        = (const float*)d_in[11];
  const int*   adj_all    = (const int*)  d_in[12];
  const float* num_w      = (const float*)d_in[13];

  float* ws    = (float*)d_ws;
  float* h     = ws;                      // [BS,D]
  float* buf1  = h    + (long)BS*D;       // hWa -> agg
  float* buf2  = buf1 + (long)BS*D;       // h_n -> hl (in-place)
  float* buf3  = buf2 + (long)BS*D;       // hq  -> h_global
  float* star  = buf3 + (long)BS*D;       // [B,D]
  float* sk    = star + B*D;              // [B,D]
  float* itemh = sk   + B*D;              // [B*LASTL,D]
  float* q     = itemh + B*LASTL*D;       // [B*LASTL,D]
  float* WaT   = q    + B*LASTL*D;        // [112,100]
  float* WqT   = WaT  + NPAD*D;
  float* WkT   = WqT  + NPAD*D;
  float* WlT   = WkT  + NPAD*D;
  float* WgT   = WlT  + NPAD*D;           // [112,200]
  float* adjP  = WgT  + NPAD*2*D;         // [B,64,52]
  float* XT    = adjP + (long)B*64*52;    // [B,112,52]

  float* out_main = (float*)d_out;              // [B,S,D]
  float* out_last = out_main + (long)BS*D;      // [B,D]

  auto gblk = [](long waves){ return (int)((waves*32 + 255)/256); };
  const long tiles_bsd = (BS/16) * (NPAD/16);        // 1600*7
  const long tiles_bd  = (B/16)  * (NPAD/16);        // 32*7
  const long tiles_q   = (B*LASTL/16) * (NPAD/16);   // 96*7

  // 0. pad/transpose all weights
  pad_weightT<<<(NPAD*D   + 255)/256, 256, 0, stream>>>(W_star_a, WaT, D);
  pad_weightT<<<(NPAD*D   + 255)/256, 256, 0, stream>>>(W_star_q, WqT, D);
  pad_weightT<<<(NPAD*D   + 255)/256, 256, 0, stream>>>(W_star_k, WkT, D);
  pad_weightT<<<(NPAD*D   + 255)/256, 256, 0, stream>>>(W_l,      WlT, D);
  pad_weightT<<<(NPAD*2*D + 255)/256, 256, 0, stream>>>(W_g,      WgT, 2*D);
  pad_adj_kernel<<<(B*64*52 + 255)/256, 256, 0, stream>>>(adj, adjP);
  // 1. h = embedding[inputs]
  gather_rows<<<(BS*D + 255)/256, 256, 0, stream>>>(embedding, inputs, h, BS*D);
  // 2. star = masked mean
  star_kernel<<<B, 128, 0, stream>>>(h, inputs, star);
  // 3. hWa = h@W_star_a (buf1), hq = h@W_star_q (buf3)  -- fused dual GEMM
  wmma_gemm_dual<D><<<gblk(tiles_bsd), 256, 0, stream>>>(h, WaT, WqT, buf1, buf3, BS);
  // 4. h_n = adj @ hWa : pad/transpose hWa then batched WMMA
  padT_kernel<<<(B*NPAD*52 + 255)/256, 256, 0, stream>>>(buf1, XT);
  adj_gemm<<<gblk((long)B*28), 256, 0, stream>>>(adjP, XT, buf2);
  // 5. sk = star @ W_star_k
  wmma_gemm<D><<<gblk(tiles_bd), 256, 0, stream>>>(star, WkT, sk, B, 0);
  // 6. gate + hl (in-place on buf2)
  gate_hl_kernel<<<gblk(BS), 256, 0, stream>>>(buf3, sk, buf2, star, buf2);
  // 7. agg (global branch neighbor softmax) -> buf1
  agg_kernel<<<gblk(BS), 256, 0, stream>>>(embedding, inputs, adj_all, num_w, buf1);
  // 8. h_global = relu(concat(h, agg) @ W_g) -> buf3
  wmma_gemm_cat<<<gblk(tiles_bsd), 256, 0, stream>>>(h, buf1, WgT, buf3, BS);
  // 9. output = s0*l2norm(hl) + s1*l2norm(hg)
  combine_kernel<<<gblk(BS), 256, 0, stream>>>(buf2, buf3, svec, out_main);
  // 10. item_h = embedding[last_items]
  gather_rows<<<(B*LASTL*D + 255)/256, 256, 0, stream>>>(embedding, last_items, itemh, B*LASTL*D);
  // 11. q = item_h @ W_l
  wmma_gemm<D><<<gblk(tiles_q), 256, 0, stream>>>(itemh, WlT, q, B*LASTL, 0);
  // 12. last_hidden
  last_kernel<<<gblk(B), 256, 0, stream>>>(embedding, itemh, q, adj_items, out_last);
}